// EventProjector_33148557591175
// MI455X (gfx1250) — compile-verified
//
#include <hip/hip_runtime.h>
#include <hip/hip_bf16.h>
#include <math.h>

#define MASK_TOKEN_ID 50264
#define Bsz 16
#define Ssz 2048
#define Hsz 1024
#define Lsz 128
#define MROWS (Bsz + Bsz * Lsz) /* 2064 rows: 16 q-rows + 2048 seq-rows */
#define EPSv 1e-8f

typedef __attribute__((ext_vector_type(2))) float v2f;
typedef __attribute__((ext_vector_type(8))) float v8f;

// ---------------------------------------------------------------------------
// Kernel 1: per-batch first occurrence of MASK_TOKEN_ID (argmax of ==mask).
// ---------------------------------------------------------------------------
__global__ void find_mask_kernel(const int* __restrict__ input_ids,
                                 int* __restrict__ mask_pos) {
    __shared__ int smin[256];
    const int b = blockIdx.x;
    const int t = threadIdx.x;
    int local = Ssz;  // sentinel: "not found"
    for (int s = t; s < Ssz; s += 256) {
        if (input_ids[(size_t)b * Ssz + s] == MASK_TOKEN_ID) { local = s; break; }
    }
    smin[t] = local;
    __syncthreads();
    for (int off = 128; off > 0; off >>= 1) {
        if (t < off) smin[t] = min(smin[t], smin[t + off]);
        __syncthreads();
    }
    if (t == 0) mask_pos[b] = (smin[0] >= Ssz) ? 0 : smin[0];  // argmax semantics
}

// ---------------------------------------------------------------------------
// Kernel 2: gather the 2064 needed rows into contiguous G[MROWS][H].
//   rows [0,16):      q_event_output[b, mask_pos[b], :]
//   rows [16,2064):   sequence_output[b, offsets[l], :]  (m = 16 + b*L + l)
// ---------------------------------------------------------------------------
__global__ void gather_rows_kernel(const float* __restrict__ q,
                                   const float* __restrict__ seq,
                                   const int* __restrict__ offsets,
                                   const int* __restrict__ mask_pos,
                                   float* __restrict__ G) {
    const int m = blockIdx.x;
    const int t = threadIdx.x;  // 256 threads, float4 each -> 1024 floats
    const float* src;
    if (m < Bsz) {
        src = q + ((size_t)m * Ssz + (size_t)mask_pos[m]) * Hsz;
    } else {
        const int mm = m - Bsz;
        const int b = mm >> 7;           // / Lsz
        const int l = mm & (Lsz - 1);    // % Lsz
        src = seq + ((size_t)b * Ssz + (size_t)offsets[l]) * Hsz;
    }
    float* dst = G + (size_t)m * Hsz;
    ((float4*)dst)[t] = ((const float4*)src)[t];
}

// ---------------------------------------------------------------------------
// Kernel 3: P[m,o] = sum_h G[m,h] * W[o,h] + bias[o]
// One wave32 per 16x16 output tile; K-loop of 4 via V_WMMA_F32_16X16X4_F32.
// A layout: lanes 0-15 rows M=0..15 (K=0,1 in the 2 VGPRs), lanes 16-31 same
// rows with K=2,3.  B layout mirrors with N across lanes.  C/D: VGPR i holds
// M=i (lanes 0-15) / M=8+i (lanes 16-31), N = lane&15.
// ---------------------------------------------------------------------------
__global__ void __launch_bounds__(32)
wmma_gemm_kernel(const float* __restrict__ G,
                 const float* __restrict__ W,
                 const float* __restrict__ bias,
                 float* __restrict__ P) {
    const int tm = blockIdx.x;       // 0..MROWS/16-1  (129 tiles)
    const int tn = blockIdx.y;       // 0..Hsz/16-1    (64 tiles)
    const int lane = threadIdx.x;    // wave32
    const int half = lane >> 4;      // 0: K=0,1   1: K=2,3
    const int l15  = lane & 15;
    const int koff = half * 2;

    const float* __restrict__ arow = G + ((size_t)(tm * 16 + l15)) * Hsz + koff;
    // B[k][n] = W[n*H + k]  (einsum 'bsh,oh->bso' => out = in @ W^T)
    const float* __restrict__ brow = W + ((size_t)(tn * 16 + l15)) * Hsz + koff;

    v8f c = {};
#pragma unroll 8
    for (int k = 0; k < Hsz; k += 4) {
        v2f a  = *(const v2f*)(arow + k);
        v2f bm = *(const v2f*)(brow + k);
        // 8 args: (neg_a, A, neg_b, B, c_mod, C, reuse_a, reuse_b)
        c = __builtin_amdgcn_wmma_f32_16x16x4_f32(
                false, a, false, bm, (short)0, c, false, false);
    }

    const int n = tn * 16 + l15;
    const float bv = bias[n];
    const int mbase = tm * 16 + half * 8;
#pragma unroll
    for (int i = 0; i < 8; ++i) {
        P[(size_t)(mbase + i) * Hsz + n] = c[i] + bv;
    }
}

// ---------------------------------------------------------------------------
// Kernel 4: per-batch cosine / exp / contrastive-ratio loss.
// Block = 256 threads = 8 waves; wave w owns l = w, w+8, ...
// ---------------------------------------------------------------------------
__global__ void finalize_kernel(const float* __restrict__ P,
                                const int* __restrict__ labels,
                                const int* __restrict__ events,
                                float* __restrict__ loss_b) {
    __shared__ float xsh[Hsz];
    __shared__ float red[8];
    __shared__ float num_sh[8];
    __shared__ float den_sh[8];

    const int b = blockIdx.x;
    const int t = threadIdx.x;
    const int lane = t & 31;
    const int wv = t >> 5;

    // stage x (projected mask row) into LDS, accumulate |x|^2
    const float* xrow = P + (size_t)b * Hsz;
    float xs = 0.f;
    for (int i = t; i < Hsz; i += 256) {
        const float v = xrow[i];
        xsh[i] = v;
        xs += v * v;
    }
    for (int off = 16; off > 0; off >>= 1) xs += __shfl_down(xs, off, 32);
    if (lane == 0) red[wv] = xs;
    __syncthreads();
    float xn2 = 0.f;
#pragma unroll
    for (int i = 0; i < 8; ++i) xn2 += red[i];
    const float xn = sqrtf(xn2);

    float num = 0.f, den = 0.f;
    for (int l = wv; l < Lsz; l += 8) {
        const float* y = P + (size_t)(Bsz + b * Lsz + l) * Hsz;
        float dot = 0.f, yn2 = 0.f;
        for (int i = lane; i < Hsz; i += 32) {
            const float yv = y[i];
            dot += yv * xsh[i];
            yn2 += yv * yv;
        }
        for (int off = 16; off > 0; off >>= 1) {
            dot += __shfl_down(dot, off, 32);
            yn2 += __shfl_down(yn2, off, 32);
        }
        if (lane == 0) {
            const float cosv = dot / fmaxf(xn * sqrtf(yn2), EPSv);
            const float e = expf(cosv);
            num += e * (float)labels[b * Lsz + l];
            den += e * (float)events[b * Lsz + l];
        }
    }
    if (lane == 0) { num_sh[wv] = num; den_sh[wv] = den; }
    __syncthreads();
    if (t == 0) {
        float N = 0.f, D = 0.f;
#pragma unroll
        for (int i = 0; i < 8; ++i) { N += num_sh[i]; D += den_sh[i]; }
        loss_b[b] = -logf(N / D);
    }
}

// ---------------------------------------------------------------------------
// Kernel 5: mean over batches -> scalar output.
// ---------------------------------------------------------------------------
__global__ void mean_kernel(const float* __restrict__ loss_b,
                            float* __restrict__ out) {
    if (threadIdx.x == 0) {
        float s = 0.f;
        for (int i = 0; i < Bsz; ++i) s += loss_b[i];
        out[0] = s / (float)Bsz;
    }
}

extern "C" void kernel_launch(void* const* d_in, const int* in_sizes, int n_in,
                              void* d_out, int out_size, void* d_ws, size_t ws_size,
                              hipStream_t stream) {
    // setup_inputs() order:
    // 0 input_ids(int B*S), 1 q_event_output(f32 B*S*H), 2 sequence_output(f32),
    // 3 events(int B*L), 4 labels(int B*L), 5 offsets(int L), 6 lengths(int B,
    // unused by the reference), 7 W(f32 H*H), 8 b(f32 H)
    const int*   input_ids = (const int*)d_in[0];
    const float* q         = (const float*)d_in[1];
    const float* seq       = (const float*)d_in[2];
    const int*   events    = (const int*)d_in[3];
    const int*   labels    = (const int*)d_in[4];
    const int*   offsets   = (const int*)d_in[5];
    const float* W         = (const float*)d_in[7];
    const float* bias      = (const float*)d_in[8];
    float* out = (float*)d_out;

    // workspace layout
    char* ws = (char*)d_ws;
    int*   mask_pos = (int*)ws;                        // 64 B
    float* loss_b   = (float*)(ws + 64);               // 64 B
    float* G        = (float*)(ws + 256);              // MROWS*H*4 = 8,454,144 B
    float* P        = (float*)(ws + 256 + (size_t)MROWS * Hsz * sizeof(float));

    find_mask_kernel<<<Bsz, 256, 0, stream>>>(input_ids, mask_pos);
    gather_rows_kernel<<<MROWS, 256, 0, stream>>>(q, seq, offsets, mask_pos, G);

    dim3 grid(MROWS / 16, Hsz / 16);  // 129 x 64 tiles, one wave32 each
    wmma_gemm_kernel<<<grid, 32, 0, stream>>>(G, W, bias, P);

    finalize_kernel<<<Bsz, 256, 0, stream>>>(P, labels, events, loss_b);
    mean_kernel<<<1, 32, 0, stream>>>(loss_b, out);
}